// Attention_39402029974027
// MI455X (gfx1250) — compile-verified
//
#include <hip/hip_runtime.h>
#include <hip/hip_bf16.h>
#include <math.h>

// Shapes from the reference
#define BB   16
#define TK   2048
#define STK  64
#define NN   512

typedef float v2f __attribute__((ext_vector_type(2)));
typedef float v8f __attribute__((ext_vector_type(8)));

// ---------------------------------------------------------------------------
// Zero-init the atomically-accumulated output regions (c_t, sent_c_t)
// ---------------------------------------------------------------------------
__global__ void zero_out_kernel(float* __restrict__ ct, float* __restrict__ sct, int n) {
    int i = blockIdx.x * blockDim.x + threadIdx.x;
    if (i < n) { ct[i] = 0.0f; sct[i] = 0.0f; }
}

// ---------------------------------------------------------------------------
// dec = S @ W^T + bias   via V_WMMA_F32_16X16X4_F32 (fp32 WMMA, wave32)
// S: [16, 512] (M=16, K=512), W: [512, 512] row-major [n][k], out: [16, 512]
// One wave (32 threads) per 16x16 output tile; gridDim.x = 32 tiles,
// gridDim.y selects word (0) vs sentence (1) operand set.
//
// A-frag (32-bit A 16x4, ISA 7.12.2): lane l, m=l&15, g=l>>4:
//   vgpr0 = A[m][k+2g], vgpr1 = A[m][k+2g+1]
// B-frag (4x16, N striped across lanes): lane l, n=l&15, g=l>>4:
//   vgpr0 = B[k+2g][n] = W[nbase+n][k+2g], vgpr1 = W[nbase+n][k+2g+1]
// C/D: vgpr j -> row (j + 8g), col nbase + (l&15)
// ---------------------------------------------------------------------------
__global__ void dec_wmma_kernel(const float* __restrict__ S0, const float* __restrict__ W0,
                                const float* __restrict__ b0, float* __restrict__ o0,
                                const float* __restrict__ S1, const float* __restrict__ W1,
                                const float* __restrict__ b1, float* __restrict__ o1) {
    const float* S    = blockIdx.y ? S1 : S0;
    const float* W    = blockIdx.y ? W1 : W0;
    const float* bias = blockIdx.y ? b1 : b0;
    float*       out  = blockIdx.y ? o1 : o0;

    const int l     = threadIdx.x & 31;
    const int m     = l & 15;           // A row for this lane
    const int g     = l >> 4;           // lane-group (selects K pair)
    const int nbase = blockIdx.x * 16;  // output column tile
    const int nn    = l & 15;           // B/C column within tile

    v8f c;
    const float bv = bias[nbase + nn];
#pragma unroll
    for (int j = 0; j < 8; ++j) c[j] = bv;

    const float* arow = S + (size_t)m * NN + 2 * g;
    const float* brow = W + (size_t)(nbase + nn) * NN + 2 * g;

#pragma unroll 8
    for (int kk = 0; kk < NN; kk += 4) {
        v2f a = *(const v2f*)(arow + kk);
        v2f b = *(const v2f*)(brow + kk);
        c = __builtin_amdgcn_wmma_f32_16x16x4_f32(
                /*neg_a=*/false, a, /*neg_b=*/false, b,
                /*c_mod=*/(short)0, c, /*reuse_a=*/false, /*reuse_b=*/false);
    }

#pragma unroll
    for (int j = 0; j < 8; ++j)
        out[(size_t)(j + 8 * g) * NN + nbase + nn] = c[j];
}

// ---------------------------------------------------------------------------
// scores[tok] = sum_n v[n] * tanh(feat[tok][n] + dec[b][n] + cov[tok]*wc[n])
// One wave per token; float4-coalesced loads; wave32 shuffle reduction.
// cov/wc == nullptr for the sentence-level path.
// ---------------------------------------------------------------------------
__global__ void scores_kernel(const float* __restrict__ feat, const float* __restrict__ dec,
                              const float* __restrict__ v, const float* __restrict__ wc,
                              const float* __restrict__ cov, float* __restrict__ scores,
                              int Tper, int ntok) {
    const int wave = threadIdx.x >> 5;
    const int lane = threadIdx.x & 31;
    const int tok  = blockIdx.x * (blockDim.x >> 5) + wave;
    if (tok >= ntok) return;                  // uniform per wave
    const int b = tok / Tper;

    const float* f = feat + (size_t)tok * NN;
    const float* d = dec + (size_t)b * NN;
    const float  cv = cov ? cov[tok] : 0.0f;

    float acc = 0.0f;
#pragma unroll
    for (int i = 0; i < 4; ++i) {
        const int n = i * 128 + lane * 4;
        float4 fv = *(const float4*)(f + n);
        float4 dv = *(const float4*)(d + n);
        float4 vv = *(const float4*)(v + n);
        float4 ev;
        if (wc) {
            float4 wv = *(const float4*)(wc + n);
            ev.x = fv.x + dv.x + cv * wv.x;  ev.y = fv.y + dv.y + cv * wv.y;
            ev.z = fv.z + dv.z + cv * wv.z;  ev.w = fv.w + dv.w + cv * wv.w;
        } else {
            ev.x = fv.x + dv.x;  ev.y = fv.y + dv.y;
            ev.z = fv.z + dv.z;  ev.w = fv.w + dv.w;
        }
        acc += vv.x * tanhf(ev.x) + vv.y * tanhf(ev.y)
             + vv.z * tanhf(ev.z) + vv.w * tanhf(ev.w);
    }
#pragma unroll
    for (int off = 16; off > 0; off >>= 1)
        acc += __shfl_xor(acc, off, 32);
    if (lane == 0) scores[tok] = acc;
}

// ---------------------------------------------------------------------------
// Sentence softmax: one block (64 threads) per batch row.
// a = exp(s-max)*mask / sum(exp(s-max)*mask)
// ---------------------------------------------------------------------------
__global__ void sent_softmax_kernel(const float* __restrict__ scores,
                                    const float* __restrict__ mask,
                                    float* __restrict__ sent_attn) {
    __shared__ float red[STK];
    const int b = blockIdx.x, t = threadIdx.x;
    const float s = scores[b * STK + t];
    red[t] = s; __syncthreads();
    for (int off = STK / 2; off > 0; off >>= 1) {
        if (t < off) red[t] = fmaxf(red[t], red[t + off]);
        __syncthreads();
    }
    const float mx = red[0]; __syncthreads();
    const float e = __expf(s - mx) * mask[b * STK + t];
    red[t] = e; __syncthreads();
    for (int off = STK / 2; off > 0; off >>= 1) {
        if (t < off) red[t] += red[t + off];
        __syncthreads();
    }
    sent_attn[b * STK + t] = e / red[0];
}

// ---------------------------------------------------------------------------
// Word softmax fused with ragged per-sentence renorm + coverage update.
// One block (256 threads) per batch row; each thread owns 8 tokens.
// ---------------------------------------------------------------------------
__global__ void word_softmax_seg_kernel(const float* __restrict__ scores,
                                        const float* __restrict__ mask,
                                        const int*   __restrict__ ends,
                                        const float* __restrict__ cov,
                                        const float* __restrict__ sent_attn,
                                        float* __restrict__ attn_out,
                                        float* __restrict__ cov_out) {
    __shared__ float red[256];
    __shared__ float ssum[STK];
    __shared__ int   es[STK];
    const int b = blockIdx.x, tid = threadIdx.x;
    if (tid < STK) { es[tid] = ends[b * STK + tid]; ssum[tid] = 0.0f; }

    float sv[8];
    float mx = -INFINITY;
#pragma unroll
    for (int i = 0; i < 8; ++i) {
        sv[i] = scores[b * TK + i * 256 + tid];
        mx = fmaxf(mx, sv[i]);
    }
    red[tid] = mx; __syncthreads();
    for (int off = 128; off > 0; off >>= 1) {
        if (tid < off) red[tid] = fmaxf(red[tid], red[tid + off]);
        __syncthreads();
    }
    mx = red[0]; __syncthreads();

    float psum = 0.0f;
#pragma unroll
    for (int i = 0; i < 8; ++i) {
        sv[i] = __expf(sv[i] - mx) * mask[b * TK + i * 256 + tid];
        psum += sv[i];
    }
    red[tid] = psum; __syncthreads();
    for (int off = 128; off > 0; off >>= 1) {
        if (tid < off) red[tid] += red[tid + off];
        __syncthreads();
    }
    const float tot = red[0]; __syncthreads();

    float a[8]; int segc[8]; bool valid[8];
#pragma unroll
    for (int i = 0; i < 8; ++i) {
        const int t = i * 256 + tid;
        a[i] = sv[i] / tot;
        attn_out[b * TK + t] = a[i];
        int cnt = 0;
        for (int j = 0; j < STK; ++j) cnt += (es[j] <= t) ? 1 : 0;   // searchsorted right
        valid[i] = (cnt < STK);
        segc[i]  = cnt < STK - 1 ? cnt : STK - 1;
        if (valid[i]) atomicAdd(&ssum[segc[i]], a[i]);
    }
    __syncthreads();
#pragma unroll
    for (int i = 0; i < 8; ++i) {
        const int t = i * 256 + tid;
        const float asw = valid[i]
            ? a[i] / ssum[segc[i]] * sent_attn[b * STK + segc[i]] : 0.0f;
        cov_out[b * TK + t] = cov[b * TK + t] + asw;
    }
}

// ---------------------------------------------------------------------------
// c[b][n] += sum_{t in chunk} a[b][t] * enc[b][t][n]
// grid (B, nchunks), 256 threads, each thread owns n and n+256.
// Accumulated via global f32 atomics (regions pre-zeroed).
// ---------------------------------------------------------------------------
__global__ void ctx_kernel(const float* __restrict__ attn, const float* __restrict__ enc,
                           float* __restrict__ ct, int T, int Tper) {
    const int b   = blockIdx.x;
    const int t0  = blockIdx.y * Tper;
    const int n0  = threadIdx.x;
    const int n1  = threadIdx.x + 256;
    float acc0 = 0.0f, acc1 = 0.0f;
    for (int t = t0; t < t0 + Tper; ++t) {
        const float at = attn[b * T + t];
        const float* row = enc + ((size_t)b * T + t) * NN;
        acc0 += at * row[n0];
        acc1 += at * row[n1];
    }
    atomicAdd(&ct[b * NN + n0], acc0);
    atomicAdd(&ct[b * NN + n1], acc1);
}

// ---------------------------------------------------------------------------
extern "C" void kernel_launch(void* const* d_in, const int* in_sizes, int n_in,
                              void* d_out, int out_size, void* d_ws, size_t ws_size,
                              hipStream_t stream) {
    const float* s_t_hat       = (const float*)d_in[0];
    const int*   enc_sent_pos  = (const int*)  d_in[1];
    const float* enc_outputs   = (const float*)d_in[2];
    const float* enc_feature   = (const float*)d_in[3];
    const float* enc_mask      = (const float*)d_in[4];
    const float* sent_s_t_hat  = (const float*)d_in[5];
    const float* sent_enc_out  = (const float*)d_in[6];
    const float* sent_enc_feat = (const float*)d_in[7];
    const float* sent_enc_mask = (const float*)d_in[8];
    const float* coverage      = (const float*)d_in[9];
    const float* W_dec         = (const float*)d_in[10];
    const float* b_dec         = (const float*)d_in[11];
    const float* v_w           = (const float*)d_in[12];
    const float* W_sent_dec    = (const float*)d_in[13];
    const float* b_sent_dec    = (const float*)d_in[14];
    const float* sent_v_w      = (const float*)d_in[15];
    const float* w_c           = (const float*)d_in[16];

    float* out        = (float*)d_out;
    float* o_ct       = out;                          // [16,512]
    float* o_attn     = out + BB * NN;                // [16,2048]
    float* o_cov      = o_attn + BB * TK;             // [16,2048]
    float* o_sct      = o_cov + BB * TK;              // [16,512]
    float* o_sattn    = o_sct + BB * NN;              // [16,64]

    float* ws          = (float*)d_ws;                // needs ~200KB
    float* ws_dec      = ws;                          // [16,512]
    float* ws_sdec     = ws_dec + BB * NN;            // [16,512]
    float* ws_scores   = ws_sdec + BB * NN;           // [16,2048]
    float* ws_sscores  = ws_scores + BB * TK;         // [16,64]

    // 1. zero the atomically-accumulated outputs
    zero_out_kernel<<<(BB * NN + 255) / 256, 256, 0, stream>>>(o_ct, o_sct, BB * NN);

    // 2. dec / sent_dec via fp32 WMMA (grid: 32 N-tiles x {word,sent}, 1 wave each)
    dec_wmma_kernel<<<dim3(NN / 16, 2), 32, 0, stream>>>(
        s_t_hat, W_dec, b_dec, ws_dec,
        sent_s_t_hat, W_sent_dec, b_sent_dec, ws_sdec);

    // 3. word scores (32768 tokens, 8 waves/block)
    scores_kernel<<<(BB * TK) / 8, 256, 0, stream>>>(
        enc_feature, ws_dec, v_w, w_c, coverage, ws_scores, TK, BB * TK);

    // 4. sentence scores (1024 tokens)
    scores_kernel<<<(BB * STK) / 8, 256, 0, stream>>>(
        sent_enc_feat, ws_sdec, sent_v_w, nullptr, nullptr, ws_sscores, STK, BB * STK);

    // 5. sentence softmax -> sent_attn_dist output
    sent_softmax_kernel<<<BB, STK, 0, stream>>>(ws_sscores, sent_enc_mask, o_sattn);

    // 6. word softmax + ragged segment renorm + coverage update
    word_softmax_seg_kernel<<<BB, 256, 0, stream>>>(
        ws_scores, enc_mask, enc_sent_pos, coverage, o_sattn, o_attn, o_cov);

    // 7. context vectors
    ctx_kernel<<<dim3(BB, 16), 256, 0, stream>>>(o_attn, enc_outputs, o_ct, TK, TK / 16);
    ctx_kernel<<<dim3(BB, 1), 256, 0, stream>>>(o_sattn, sent_enc_out, o_sct, STK, STK);
}